// ScatterConnection_69758858822260
// MI455X (gfx1250) — compile-verified
//
#include <hip/hip_runtime.h>
#include <stdint.h>

// Problem constants (from reference)
#define BATCH 32
#define MDIM  512
#define NDIM  128
#define HDIM  128
#define WDIM  128
#define HW    (HDIM * WDIM)          // 16384

// Emit-kernel tiling: block handles (b, 1024 p's), loops over n in chunks of 8.
#define TP        1024               // p positions per block
#define TN        8                  // n rows per TDM chunk
#define NCHUNK    (NDIM / TN)        // 16
#define CHUNK_FLOATS (TN * TP)       // 8192 floats = 32 KB per buffer
#define EMIT_THREADS 256

typedef uint32_t u32;
typedef uint64_t u64;
typedef unsigned int v4u __attribute__((ext_vector_type(4)));
typedef int          v8i __attribute__((ext_vector_type(8)));
typedef int          v4i __attribute__((ext_vector_type(4)));
typedef float        v4f __attribute__((ext_vector_type(4)));

// Explicit LDS (addrspace 3) pointer type: forces ds_store_b128 lowering.
typedef volatile __attribute__((address_space(3))) v4f* lds_v4f_p;

// ---------------------------------------------------------------------------
// Kernel 1: inv[b*HW + p] = -1 for all p  (2 MB, stays L2-resident)
// ---------------------------------------------------------------------------
__global__ __launch_bounds__(256) void scat_init_inv(int* __restrict__ inv) {
    const int i = (blockIdx.x * 256 + threadIdx.x) * 4;
    *reinterpret_cast<int4*>(inv + i) = make_int4(-1, -1, -1, -1);
}

// ---------------------------------------------------------------------------
// Kernel 2: inv[b*HW + y*W + x] = m   (B*M = 16384 threads)
// location is (B, M, 2) int32 as (y, x)
// ---------------------------------------------------------------------------
__global__ __launch_bounds__(256) void scat_build_inv(const int* __restrict__ loc,
                                                      int* __restrict__ inv) {
    const int i = blockIdx.x * 256 + threadIdx.x;   // 0 .. B*M-1
    const int b = i >> 9;                           // /M (512)
    const int m = i & (MDIM - 1);
    const int y = loc[2 * i + 0];
    const int x = loc[2 * i + 1];
    inv[b * HW + y * WDIM + x] = m;
}

// ---------------------------------------------------------------------------
// TDM descriptor issue: store a TN x TP f32 tile from LDS into out with
// row stride HW (2D D#, data_size = 4 bytes). 6-arg builtin on this
// toolchain: (g0 v4u, g1 v8i, g2 v4i, g3 v4i, v8i, i32 cpol).
// All inputs are block-uniform.
// ---------------------------------------------------------------------------
__device__ __forceinline__ void tdm_store_tile(u32 lds_byte_addr, const float* gptr) {
    const u64 ga = (u64)(uintptr_t)gptr;

    // --- D# group 0 (128 bits) ---
    v4u g0;
    g0.x = 1u;                                   // count=1, is_restore=0, gather=0
    g0.y = lds_byte_addr;                        // lds_addr[31:0]
    g0.z = (u32)(ga & 0xFFFFFFFFu);              // global_addr[31:0]  (bits 95:64)
    g0.w = (u32)((ga >> 32) & 0x01FFFFFFu)       // global_addr[56:32]
         | (2u << 30);                           // type = 2 ("image")

    // --- D# group 1 (256 bits) ---
    const u32 tdim0 = HW;                        // tensor_dim0 (elements)
    const u32 tdim1 = NDIM;                      // tensor_dim1
    const u32 s0    = HW;                        // tensor_dim0_stride (elements)
    v8i g1;
    g1[0] = (int)(2u << 16);                     // wg_mask=0, data_size=2 (4B), no flags
    g1[1] = (int)((tdim0 & 0xFFFFu) << 16);      // [63:48] = tensor_dim0 lo16
    g1[2] = (int)((tdim0 >> 16) | ((tdim1 & 0xFFFFu) << 16)); // dim0 hi16 | dim1 lo16
    g1[3] = (int)((tdim1 >> 16) | ((u32)TP << 16));           // dim1 hi16 | tile_dim0
    g1[4] = (int)((u32)TN);                      // tile_dim1 = TN, tile_dim2 = 0 (2D)
    g1[5] = (int)(s0 & 0xFFFFFFFFu);             // stride0[31:0]
    g1[6] = 0;                                   // stride0[47:32]=0 | stride1[15:0]=0
    g1[7] = 0;                                   // stride1[47:16]=0

    // --- D# groups 2/3 + trailing group: unused for 2D tile (tile_dim2 == 0)
    v4i gz4 = {0, 0, 0, 0};
    v8i gz8 = {0, 0, 0, 0, 0, 0, 0, 0};

    __builtin_amdgcn_tensor_store_from_lds(g0, g1, gz4, gz4, gz8, /*cpol=*/0);
}

// ---------------------------------------------------------------------------
// Kernel 3 (main): coalesced emit via LDS compose + TDM DMA stores.
// Grid: B * (HW/TP) = 512 blocks, 256 threads (8 waves).
// Each thread owns 4 consecutive p's (inv entries held in registers across
// the whole n-loop); per TN-chunk it composes float4 rows into LDS; wave 0
// launches TENSOR_STORE_FROM_LDS, double-buffered on TENSORcnt.
// ---------------------------------------------------------------------------
__global__ __launch_bounds__(EMIT_THREADS) void scat_emit(const float* __restrict__ x,
                                                          const int* __restrict__ inv,
                                                          float* __restrict__ out) {
    __shared__ __attribute__((aligned(16))) float tile[2 * CHUNK_FLOATS]; // 64 KB

    const int tid  = threadIdx.x;
    const int b    = blockIdx.x >> 4;            // 16 p-tiles per batch
    const int p0   = (blockIdx.x & 15) * TP;
    const int pofs = tid * 4;

    // LDS byte offset of the tile: generic (flat) LDS addresses carry the
    // wave-relative LDS offset in their low 32 bits. The ptrtoint also makes
    // the array address escape, so the clobbers below alias it.
    const u32 ldsBase = (u32)(uintptr_t)(void*)&tile[0];
    asm volatile("" :: "v"((void*)&tile[0]) : "memory");   // capture LDS array

    // Inverse-map entries for this thread's 4 p's (constant across n-loop).
    const int4 mv = *reinterpret_cast<const int4*>(inv + b * HW + p0 + pofs);
    const float* xb = x + (size_t)b * (MDIM * NDIM);
    const float* r0 = (mv.x >= 0) ? xb + (size_t)mv.x * NDIM : nullptr;
    const float* r1 = (mv.y >= 0) ? xb + (size_t)mv.y * NDIM : nullptr;
    const float* r2 = (mv.z >= 0) ? xb + (size_t)mv.z * NDIM : nullptr;
    const float* r3 = (mv.w >= 0) ? xb + (size_t)mv.w * NDIM : nullptr;

    // Warm the x rows we will stream (each row is read exactly once, by us).
    if (r0) __builtin_prefetch(r0, 0, 3);
    if (r1) __builtin_prefetch(r1, 0, 3);
    if (r2) __builtin_prefetch(r2, 0, 3);
    if (r3) __builtin_prefetch(r3, 0, 3);

    for (int c = 0; c < NCHUNK; ++c) {
        const int buf = c & 1;
        // Before overwriting this buffer, make sure the TDM store issued two
        // chunks ago (same buffer) has drained: allow at most 1 outstanding.
        if (c >= 2 && tid < 32) {
            __builtin_amdgcn_s_wait_tensorcnt(1);
        }
        __syncthreads();                          // others wait for wave0's drain

        const int n0 = c * TN;
        // Destination as an explicit addrspace(3) pointer built from the
        // 32-bit LDS offset -> guaranteed ds_store_b128 (not flat_store).
        // Volatile: the reader is the TDM engine, invisible to the compiler.
        const u32 dstOff = ldsBase
                         + (u32)(buf * CHUNK_FLOATS + pofs) * (u32)sizeof(float);
        lds_v4f_p dst = (lds_v4f_p)dstOff;
        #pragma unroll
        for (int j = 0; j < TN; ++j) {
            const int n = n0 + j;
            v4f v;
            v.x = r0 ? r0[n] : 0.0f;
            v.y = r1 ? r1[n] : 0.0f;
            v.z = r2 ? r2[n] : 0.0f;
            v.w = r3 ? r3[n] : 0.0f;
            dst[j * (TP / 4)] = v;                // ds_store_b128, no bank conflicts
        }
        asm volatile("" ::: "memory");
        __syncthreads();                          // all LDS writes complete (DScnt)

        if (tid < 32) {                           // one wave issues the DMA
            tdm_store_tile(ldsBase + (u32)(buf * CHUNK_FLOATS * sizeof(float)),
                           out + ((size_t)(b * NDIM + n0) * HW + p0));
        }
    }
    // S_ENDPGM performs an implicit wait-idle (covers TENSORcnt tail).
}

// ---------------------------------------------------------------------------
// Fallback path (only if d_ws cannot hold the 2 MB inverse map):
// zero-fill with non-temporal b128 stores, then direct (uncoalesced) scatter.
// ---------------------------------------------------------------------------
__global__ __launch_bounds__(256) void scat_zero_out(float* __restrict__ out) {
    const size_t i = ((size_t)blockIdx.x * 256 + threadIdx.x) * 4;
    v4f z = {0.0f, 0.0f, 0.0f, 0.0f};
    __builtin_nontemporal_store(z, reinterpret_cast<v4f*>(out + i));
}

__global__ __launch_bounds__(256) void scat_direct(const float* __restrict__ x,
                                                   const int* __restrict__ loc,
                                                   float* __restrict__ out) {
    const int i  = blockIdx.x * 256 + threadIdx.x; // 0 .. B*M*N-1
    const int n  = i & (NDIM - 1);
    const int bm = i >> 7;                         // b*M + m
    const int b  = bm >> 9;
    const int y  = loc[2 * bm + 0];
    const int xw = loc[2 * bm + 1];
    const size_t o = ((size_t)(b * NDIM + n)) * HW + (size_t)y * WDIM + xw;
    out[o] = x[i];
}

// ---------------------------------------------------------------------------
extern "C" void kernel_launch(void* const* d_in, const int* in_sizes, int n_in,
                              void* d_out, int out_size, void* d_ws, size_t ws_size,
                              hipStream_t stream) {
    const float* x   = (const float*)d_in[0];
    const int*   loc = (const int*)d_in[1];   // (B, M, 2) int32 (y, x)
    float*       out = (float*)d_out;         // (B, N, H, W) f32
    (void)in_sizes; (void)n_in; (void)out_size;

    const size_t inv_bytes = (size_t)BATCH * HW * sizeof(int);   // 2 MB
    if (ws_size >= inv_bytes) {
        int* inv = (int*)d_ws;
        // 1) inv = -1              (B*HW/4 int4 writes)
        scat_init_inv<<<(BATCH * HW) / (4 * 256), 256, 0, stream>>>(inv);
        // 2) scatter m into inv    (B*M threads)
        scat_build_inv<<<(BATCH * MDIM) / 256, 256, 0, stream>>>(loc, inv);
        // 3) coalesced emit via LDS + TDM stores (512 blocks)
        scat_emit<<<BATCH * (HW / TP), EMIT_THREADS, 0, stream>>>(x, inv, out);
    } else {
        // Fallback: zero-fill (NT streaming) then direct scatter.
        scat_zero_out<<<(int)(((size_t)BATCH * NDIM * HW) / (4 * 256)), 256, 0, stream>>>(out);
        scat_direct<<<(BATCH * MDIM * NDIM) / 256, 256, 0, stream>>>(x, loc, out);
    }
}